// DQN_LSTM_88227218194755
// MI455X (gfx1250) — compile-verified
//
#include <hip/hip_runtime.h>
#include <math.h>

// Problem constants (match reference)
#define NROWS    65536
#define IN_DIM   28
#define HID      128
#define GATES    512
#define D1       64
#define OUT_DIM  4
#define EPS      1e-5f

#define LDS_STRIDE 132          // 128 + 4 pad: A-frag rows hit distinct bank groups
#define ROWS_PER_WG 64
#define THREADS 64              // 2 waves (wave32), 32 rows per wave (2 M-tiles)

typedef float v2f __attribute__((ext_vector_type(2)));
typedef float v8f __attribute__((ext_vector_type(8)));

// D = A(16x4 f32) * B(4x16 f32) + C(16x16 f32) ; exact f32 matrix op on CDNA5
static __device__ __forceinline__ v8f wmma4(v2f a, v2f b, v8f c) {
  return __builtin_amdgcn_wmma_f32_16x16x4_f32(
      /*neg_a=*/false, a, /*neg_b=*/false, b,
      /*c_mod=*/(short)0, c, /*reuse_a=*/false, /*reuse_b=*/false);
}

static __device__ __forceinline__ v2f ld2(const float* p) {
  return *(const v2f*)p;   // 8B-aligned at every call site -> b64 load
}
static __device__ __forceinline__ float sigm(float x) { return 1.f / (1.f + expf(-x)); }

// LayerNorm(128) + ReLU on one 16x128 slab (8 C-tiles) held in registers.
// Row r lives in VGPR v = r%8 across the 16 lanes of half r/8, so shfl_xor
// with masks 1,2,4,8 (stays inside the 16-lane half) yields full row sums.
static __device__ __forceinline__ void ln_relu(v8f* acc, const float* g, const float* bt,
                                               int ln16) {
  float mu[8], rs[8];
#pragma unroll
  for (int v = 0; v < 8; ++v) {
    float s = 0.f, ss = 0.f;
#pragma unroll
    for (int t = 0; t < 8; ++t) { float x = acc[t][v]; s += x; ss += x * x; }
#pragma unroll
    for (int m = 1; m < 16; m <<= 1) { s += __shfl_xor(s, m, 32); ss += __shfl_xor(ss, m, 32); }
    float mean = s * (1.f / 128.f);
    float var  = ss * (1.f / 128.f) - mean * mean;
    mu[v] = mean;
    rs[v] = rsqrtf(var + EPS);
  }
#pragma unroll
  for (int t = 0; t < 8; ++t) {
    float gg = g[t * 16 + ln16], bb = bt[t * 16 + ln16];
#pragma unroll
    for (int v = 0; v < 8; ++v) {
      float y = (acc[t][v] - mu[v]) * rs[v] * gg + bb;
      acc[t][v] = fmaxf(y, 0.f);
    }
  }
}

__global__ __launch_bounds__(THREADS)
void dqn_lstm_fused(const float* __restrict__ x,  const float* __restrict__ h0,
                    const float* __restrict__ c0,
                    const float* __restrict__ We1, const float* __restrict__ be1,
                    const float* __restrict__ g1,  const float* __restrict__ bt1,
                    const float* __restrict__ We2, const float* __restrict__ be2,
                    const float* __restrict__ g2,  const float* __restrict__ bt2,
                    const float* __restrict__ Wih, const float* __restrict__ Whh,
                    const float* __restrict__ bih, const float* __restrict__ bhh,
                    const float* __restrict__ Wd1, const float* __restrict__ bd1,
                    const float* __restrict__ Wd2, const float* __restrict__ bd2,
                    float* __restrict__ q_out, float* __restrict__ h_out,
                    float* __restrict__ c_out) {
  __shared__ float sAct[ROWS_PER_WG * LDS_STRIDE];

  const int tid  = threadIdx.x;
  const int lane = tid & 31;
  const int wave = tid >> 5;
  const int ln16 = lane & 15;
  const int half = lane >> 4;          // 0: K=k0,k0+1 / rows 0-7 ; 1: K=k0+2,k0+3 / rows 8-15
  const int kofs = half * 2;
  const int m0    = wave * 32;                         // wave's 32 rows inside WG tile
  const int gRow0 = blockIdx.x * ROWS_PER_WG + m0;     // global base row of this wave
  // Per M-sub-tile (u = 0,1) addressing
  const int aRow0 = gRow0 + ln16,       aRow1 = gRow0 + 16 + ln16;   // A-frag rows (global)
  const int lRow0 = m0 + ln16,          lRow1 = m0 + 16 + ln16;      // A-frag rows (LDS)

  v8f acc0[8], acc1[8];
  const v8f vz = {0.f, 0.f, 0.f, 0.f, 0.f, 0.f, 0.f, 0.f};

  // ---------------- encoder 1: e1 = relu(LN(x @ We1^T + be1)) ----------------
  {
    const float* ax0 = x + aRow0 * IN_DIM + kofs;      // K = 28 = 7*4, no padding needed
    const float* ax1 = x + aRow1 * IN_DIM + kofs;
    for (int t = 0; t < 8; ++t) {
      v8f c0a = vz, c1a = vz;
      const float* bw = We1 + (t * 16 + ln16) * IN_DIM + kofs;  // B[k][n] = We1[n][k]
#pragma unroll
      for (int k0 = 0; k0 < IN_DIM; k0 += 4) {
        v2f b = ld2(bw + k0);                                   // one B frag, two A tiles
        c0a = wmma4(ld2(ax0 + k0), b, c0a);
        c1a = wmma4(ld2(ax1 + k0), b, c1a);
      }
      float bias = be1[t * 16 + ln16];
#pragma unroll
      for (int v = 0; v < 8; ++v) { c0a[v] += bias; c1a[v] += bias; }
      acc0[t] = c0a; acc1[t] = c1a;
    }
    ln_relu(acc0, g1, bt1, ln16);
    ln_relu(acc1, g1, bt1, ln16);
  }
  for (int t = 0; t < 8; ++t)
#pragma unroll
    for (int v = 0; v < 8; ++v) {
      sAct[(m0 +      v + half * 8) * LDS_STRIDE + t * 16 + ln16] = acc0[t][v];
      sAct[(m0 + 16 + v + half * 8) * LDS_STRIDE + t * 16 + ln16] = acc1[t][v];
    }
  __syncthreads();

  // ---------------- encoder 2: e2 = relu(LN(e1 @ We2^T + be2)) ----------------
  {
    const float* al0 = sAct + lRow0 * LDS_STRIDE + kofs;
    const float* al1 = sAct + lRow1 * LDS_STRIDE + kofs;
    for (int t = 0; t < 8; ++t) {
      v8f c0a = vz, c1a = vz;
      const float* bw = We2 + (t * 16 + ln16) * HID + kofs;
      for (int k0 = 0; k0 < HID; k0 += 4) {
        v2f b = ld2(bw + k0);
        c0a = wmma4(ld2(al0 + k0), b, c0a);
        c1a = wmma4(ld2(al1 + k0), b, c1a);
      }
      float bias = be2[t * 16 + ln16];
#pragma unroll
      for (int v = 0; v < 8; ++v) { c0a[v] += bias; c1a[v] += bias; }
      acc0[t] = c0a; acc1[t] = c1a;
    }
    ln_relu(acc0, g2, bt2, ln16);
    ln_relu(acc1, g2, bt2, ln16);
  }
  __syncthreads();
  for (int t = 0; t < 8; ++t)
#pragma unroll
    for (int v = 0; v < 8; ++v) {
      sAct[(m0 +      v + half * 8) * LDS_STRIDE + t * 16 + ln16] = acc0[t][v];
      sAct[(m0 + 16 + v + half * 8) * LDS_STRIDE + t * 16 + ln16] = acc1[t][v];
    }
  __syncthreads();

  // ---------------- LSTM cell: gates = e2@Wih^T + h0@Whh^T + bih + bhh -------
  // Per 16-col tile compute all 4 gates for both M-sub-tiles (8 accumulators)
  // so each loaded weight fragment feeds two WMMAs.
  {
    const float* al0 = sAct + lRow0 * LDS_STRIDE + kofs;   // e2 (A operand, LDS)
    const float* al1 = sAct + lRow1 * LDS_STRIDE + kofs;
    const float* ah0 = h0 + aRow0 * HID + kofs;            // h0 (A operand, global)
    const float* ah1 = h0 + aRow1 * HID + kofs;
    for (int t = 0; t < 8; ++t) {
      v8f ai0 = vz, af0 = vz, ag0 = vz, ao0 = vz;
      v8f ai1 = vz, af1 = vz, ag1 = vz, ao1 = vz;
      const int nb = t * 16 + ln16;
      const float* bi = Wih + (0 * HID + nb) * HID + kofs;
      const float* bf = Wih + (1 * HID + nb) * HID + kofs;
      const float* bg = Wih + (2 * HID + nb) * HID + kofs;
      const float* bo = Wih + (3 * HID + nb) * HID + kofs;
      for (int k0 = 0; k0 < HID; k0 += 4) {
        v2f a0 = ld2(al0 + k0), a1 = ld2(al1 + k0);
        v2f b;
        b = ld2(bi + k0); ai0 = wmma4(a0, b, ai0); ai1 = wmma4(a1, b, ai1);
        b = ld2(bf + k0); af0 = wmma4(a0, b, af0); af1 = wmma4(a1, b, af1);
        b = ld2(bg + k0); ag0 = wmma4(a0, b, ag0); ag1 = wmma4(a1, b, ag1);
        b = ld2(bo + k0); ao0 = wmma4(a0, b, ao0); ao1 = wmma4(a1, b, ao1);
      }
      const float* ci = Whh + (0 * HID + nb) * HID + kofs;
      const float* cf = Whh + (1 * HID + nb) * HID + kofs;
      const float* cg = Whh + (2 * HID + nb) * HID + kofs;
      const float* co = Whh + (3 * HID + nb) * HID + kofs;
      for (int k0 = 0; k0 < HID; k0 += 4) {
        v2f a0 = ld2(ah0 + k0), a1 = ld2(ah1 + k0);
        v2f b;
        b = ld2(ci + k0); ai0 = wmma4(a0, b, ai0); ai1 = wmma4(a1, b, ai1);
        b = ld2(cf + k0); af0 = wmma4(a0, b, af0); af1 = wmma4(a1, b, af1);
        b = ld2(cg + k0); ag0 = wmma4(a0, b, ag0); ag1 = wmma4(a1, b, ag1);
        b = ld2(co + k0); ao0 = wmma4(a0, b, ao0); ao1 = wmma4(a1, b, ao1);
      }
      const int col = t * 16 + ln16;
      const float bI = bih[0 * HID + col] + bhh[0 * HID + col];
      const float bF = bih[1 * HID + col] + bhh[1 * HID + col];
      const float bG = bih[2 * HID + col] + bhh[2 * HID + col];
      const float bO = bih[3 * HID + col] + bhh[3 * HID + col];
#pragma unroll
      for (int v = 0; v < 8; ++v) {
        {  // sub-tile u = 0
          const int row = gRow0 + v + half * 8;
          float iv = sigm(ai0[v] + bI);
          float fv = sigm(af0[v] + bF);
          float gv = tanhf(ag0[v] + bG);
          float ov = sigm(ao0[v] + bO);
          float cv = fv * c0[row * HID + col] + iv * gv;
          float hv = ov * tanhf(cv);
          c_out[row * HID + col] = cv;
          h_out[row * HID + col] = hv;
        }
        {  // sub-tile u = 1
          const int row = gRow0 + 16 + v + half * 8;
          float iv = sigm(ai1[v] + bI);
          float fv = sigm(af1[v] + bF);
          float gv = tanhf(ag1[v] + bG);
          float ov = sigm(ao1[v] + bO);
          float cv = fv * c0[row * HID + col] + iv * gv;
          float hv = ov * tanhf(cv);
          c_out[row * HID + col] = cv;
          h_out[row * HID + col] = hv;
        }
      }
    }
  }

  // Decoder re-reads h_n straight from h_out (same-wave store->load to the same
  // address is kept in order per ISA; drain STOREcnt explicitly before reading).
  asm volatile("s_wait_storecnt 0x0" ::: "memory");

  // ---------------- decoder 1: d1 = relu(h_n @ Wd1^T + bd1), N=64 ------------
  {
    const float* ahn0 = h_out + aRow0 * HID + kofs;
    const float* ahn1 = h_out + aRow1 * HID + kofs;
    for (int t = 0; t < 4; ++t) {
      v8f c0a = vz, c1a = vz;
      const float* bw = Wd1 + (t * 16 + ln16) * HID + kofs;
      for (int k0 = 0; k0 < HID; k0 += 4) {
        v2f b = ld2(bw + k0);
        c0a = wmma4(ld2(ahn0 + k0), b, c0a);
        c1a = wmma4(ld2(ahn1 + k0), b, c1a);
      }
      float b = bd1[t * 16 + ln16];
#pragma unroll
      for (int v = 0; v < 8; ++v) {
        c0a[v] = fmaxf(c0a[v] + b, 0.f);
        c1a[v] = fmaxf(c1a[v] + b, 0.f);
      }
      acc0[t] = c0a; acc1[t] = c1a;
    }
  }
  __syncthreads();
  for (int t = 0; t < 4; ++t)
#pragma unroll
    for (int v = 0; v < 8; ++v) {
      sAct[(m0 +      v + half * 8) * LDS_STRIDE + t * 16 + ln16] = acc0[t][v];
      sAct[(m0 + 16 + v + half * 8) * LDS_STRIDE + t * 16 + ln16] = acc1[t][v];
    }
  __syncthreads();

  // ---------------- decoder 2: q = d1 @ Wd2^T + bd2, N=4 (padded to 16) ------
  {
    const float* al0 = sAct + lRow0 * LDS_STRIDE + kofs;
    const float* al1 = sAct + lRow1 * LDS_STRIDE + kofs;
    const float scale = (ln16 < OUT_DIM) ? 1.f : 0.f;          // branch-free N pad
    const float* bw = Wd2 + (ln16 & 3) * D1 + kofs;            // always-valid addr
    v8f c0a = vz, c1a = vz;
    for (int k0 = 0; k0 < D1; k0 += 4) {
      v2f b = ld2(bw + k0);
      b[0] *= scale; b[1] *= scale;
      c0a = wmma4(ld2(al0 + k0), b, c0a);
      c1a = wmma4(ld2(al1 + k0), b, c1a);
    }
    if (ln16 < OUT_DIM) {
      const float bq = bd2[ln16];
#pragma unroll
      for (int v = 0; v < 8; ++v) {
        q_out[(gRow0 +      v + half * 8) * OUT_DIM + ln16] = c0a[v] + bq;
        q_out[(gRow0 + 16 + v + half * 8) * OUT_DIM + ln16] = c1a[v] + bq;
      }
    }
  }
}

extern "C" void kernel_launch(void* const* d_in, const int* in_sizes, int n_in,
                              void* d_out, int out_size, void* d_ws, size_t ws_size,
                              hipStream_t stream) {
  (void)in_sizes; (void)n_in; (void)d_ws; (void)ws_size; (void)out_size;
  const float* x   = (const float*)d_in[0];
  const float* h0  = (const float*)d_in[1];   // [1,B,HID] == [B,HID]
  const float* c0  = (const float*)d_in[2];
  const float* We1 = (const float*)d_in[3];
  const float* be1 = (const float*)d_in[4];
  const float* g1  = (const float*)d_in[5];
  const float* bt1 = (const float*)d_in[6];
  const float* We2 = (const float*)d_in[7];
  const float* be2 = (const float*)d_in[8];
  const float* g2  = (const float*)d_in[9];
  const float* bt2 = (const float*)d_in[10];
  const float* Wih = (const float*)d_in[11];
  const float* Whh = (const float*)d_in[12];
  const float* bih = (const float*)d_in[13];
  const float* bhh = (const float*)d_in[14];
  const float* Wd1 = (const float*)d_in[15];
  const float* bd1 = (const float*)d_in[16];
  const float* Wd2 = (const float*)d_in[17];
  const float* bd2 = (const float*)d_in[18];

  float* q_out = (float*)d_out;                       // [B,4]
  float* h_out = q_out + (size_t)NROWS * OUT_DIM;     // [1,B,128]
  float* c_out = h_out + (size_t)NROWS * HID;         // [1,B,128]

  dim3 grid(NROWS / ROWS_PER_WG);   // 1024
  dim3 block(THREADS);              // 64 = 2 waves, 32 rows/wave
  hipLaunchKernelGGL(dqn_lstm_fused, grid, block, 0, stream,
                     x, h0, c0, We1, be1, g1, bt1, We2, be2, g2, bt2,
                     Wih, Whh, bih, bhh, Wd1, bd1, Wd2, bd2,
                     q_out, h_out, c_out);
}